// unifiedloss_63350767616723
// MI455X (gfx1250) — compile-verified
//
#include <hip/hip_runtime.h>
#include <math.h>

typedef float v2f __attribute__((ext_vector_type(2)));
typedef float v8f __attribute__((ext_vector_type(8)));

#define EPSF 1e-7f

__device__ __forceinline__ float sigclip(float x) {
    float p = 1.0f / (1.0f + __expf(-x));
    return fminf(fmaxf(p, EPSF), 1.0f - EPSF);
}

// ---------------------------------------------------------------------------
// Kernel 1: grid-stride masked partial sums + WMMA block reduction.
// Quantities: 0:n  1:T=Σmf*t1s  2:P0=Σmf*p0  3:P1=Σmf*p1  4:Q0=Σmf*t1s*p0
//             5:Q1=Σmf*t1s*p1
// Each block writes 6 partial sums to part[q*nb + blockIdx.x].
// ---------------------------------------------------------------------------
__global__ __launch_bounds__(256) void ul_partial(const float* __restrict__ pred,
                                                  const float* __restrict__ tgt,
                                                  float* __restrict__ part,
                                                  int nb) {
    constexpr int H = 1024, W = 1024, B = 8;
    constexpr int GW = W / 4;            // float4 groups per row
    constexpr int NG = B * H * GW;       // total groups
    __shared__ float sd[6 * 256];

    const int tid = threadIdx.x;
    float acc[6] = {0.f, 0.f, 0.f, 0.f, 0.f, 0.f};

    for (int i = blockIdx.x * 256 + tid; i < NG; i += nb * 256) {
        int b   = i / (H * GW);
        int rem = i - b * H * GW;
        int y   = rem / GW;
        int x0  = (rem - y * GW) * 4;

        const size_t img = (size_t)H * W;
        const float* p0r = pred + (size_t)(2 * b) * img + (size_t)y * W + x0;
        const float* p1r = p0r + img;
        const float* gpl = tgt + (size_t)(2 * b + 1) * img;  // foreground channel

        const int ym = (y > 0) ? y - 1 : 0;
        const int yp = (y < H - 1) ? y + 1 : y;
        const int xl = (x0 > 0) ? x0 - 1 : 0;
        const int xr = (x0 + 4 < W) ? x0 + 4 : W - 1;

        float4 z0 = *reinterpret_cast<const float4*>(p0r);
        float4 z1 = *reinterpret_cast<const float4*>(p1r);

        float mn[4] = {1e30f, 1e30f, 1e30f, 1e30f};
        float mx[4] = {-1e30f, -1e30f, -1e30f, -1e30f};
        float4 gc = {0.f, 0.f, 0.f, 0.f};

        const int rows[3] = {ym, y, yp};
#pragma unroll
        for (int rr = 0; rr < 3; ++rr) {
            const float* gr = gpl + (size_t)rows[rr] * W;
            float4 v = *reinterpret_cast<const float4*>(gr + x0);
            float a[6] = {gr[xl], v.x, v.y, v.z, v.w, gr[xr]};
            if (rr == 1) gc = v;
#pragma unroll
            for (int k = 0; k < 4; ++k) {
                float lo = fminf(a[k], fminf(a[k + 1], a[k + 2]));
                float hi = fmaxf(a[k], fmaxf(a[k + 1], a[k + 2]));
                mn[k] = fminf(mn[k], lo);
                mx[k] = fmaxf(mx[k], hi);
            }
        }

        float gg[4]  = {gc.x, gc.y, gc.z, gc.w};
        float zz0[4] = {z0.x, z0.y, z0.z, z0.w};
        float zz1[4] = {z1.x, z1.y, z1.z, z1.w};
#pragma unroll
        for (int k = 0; k < 4; ++k) {
            float mf = (mn[k] == mx[k]) ? 1.0f : 0.0f;  // non-border one-hot pixel
            float ts = 0.9f * gg[k] + 0.05f;            // smoothed foreground target
            float p0 = sigclip(zz0[k]);
            float p1 = sigclip(zz1[k]);
            acc[0] += mf;
            acc[1] += mf * ts;
            acc[2] += mf * p0;
            acc[3] += mf * p1;
            acc[4] += mf * ts * p0;
            acc[5] += mf * ts * p1;
        }
    }

#pragma unroll
    for (int q = 0; q < 6; ++q) sd[q * 256 + tid] = acc[q];
    __syncthreads();

    // Wave 0 reduces 6x256 LDS values with V_WMMA_F32_16X16X4_F32, B = ones:
    // D[m,n] = sum_k A[m,k]  (row sums, f32 precision), chained over 4 tiles.
    if (tid < 32) {
        const int lane = tid;
        const int off  = ((lane & 15) << 2) | ((lane >> 4) << 1);  // A[m,k] slot
        v2f bones;
        bones[0] = 1.0f;
        bones[1] = 1.0f;
#pragma unroll
        for (int q = 0; q < 6; ++q) {
            v8f c = {};
#pragma unroll
            for (int t = 0; t < 4; ++t) {
                int base = q * 256 + t * 64 + off;
                v2f a;
                a[0] = sd[base];
                a[1] = sd[base + 1];
                c = __builtin_amdgcn_wmma_f32_16x16x4_f32(
                        false, a, false, bones, (short)0, c, false, false);
            }
            float s = c[0] + c[1] + c[2] + c[3] + c[4] + c[5] + c[6] + c[7];
            s += __shfl_xor(s, 16, 32);  // rows 0-7 half + rows 8-15 half
            if (lane == 0) part[q * nb + blockIdx.x] = s;
        }
    }
}

// ---------------------------------------------------------------------------
// Kernel 2: one wave reduces the per-block partials (WMMA chain, fixed order,
// deterministic) and evaluates the focal-Tversky formula.
// ---------------------------------------------------------------------------
__global__ __launch_bounds__(32) void ul_final(const float* __restrict__ part,
                                               float* __restrict__ out,
                                               int nb) {
    const int lane = threadIdx.x;
    const int off  = ((lane & 15) << 2) | ((lane >> 4) << 1);
    v2f bones;
    bones[0] = 1.0f;
    bones[1] = 1.0f;

    float tot[6];
#pragma unroll
    for (int q = 0; q < 6; ++q) {
        v8f c = {};
        for (int t = 0; t < nb / 64; ++t) {
            int base = q * nb + t * 64 + off;
            v2f a;
            a[0] = part[base];
            a[1] = part[base + 1];
            c = __builtin_amdgcn_wmma_f32_16x16x4_f32(
                    false, a, false, bones, (short)0, c, false, false);
        }
        float s = c[0] + c[1] + c[2] + c[3] + c[4] + c[5] + c[6] + c[7];
        s += __shfl_xor(s, 16, 32);
        tot[q] = s;
    }

    if (lane == 0) {
        const float n  = tot[0], T  = tot[1];
        const float P0 = tot[2], P1 = tot[3];
        const float Q0 = tot[4], Q1 = tot[5];

        // class 0: t0s = 1 - t1s
        float tp0 = P0 - Q0;
        float fn0 = (n - T) - tp0;
        float fp0 = Q0;
        // class 1
        float tp1 = Q1;
        float fn1 = T - Q1;
        float fp1 = P1 - Q1;

        const float e = 1e-7f, DLT = 0.6f;
        float d0 = (tp0 + e) / (tp0 + DLT * fn0 + (1.0f - DLT) * fp0 + e);
        float d1 = (tp1 + e) / (tp1 + DLT * fn1 + (1.0f - DLT) * fp1 + e);

        // (1-d1) * (1-d1)^(-0.5) == sqrt(1-d1); WEIGHT == 1.0 -> loss == ftl
        float ftl = 0.5f * ((1.0f - d0) + sqrtf(fmaxf(1.0f - d1, 0.0f)));
        out[0] = ftl;
    }
}

extern "C" void kernel_launch(void* const* d_in, const int* in_sizes, int n_in,
                              void* d_out, int out_size, void* d_ws, size_t ws_size,
                              hipStream_t stream) {
    (void)in_sizes; (void)n_in; (void)out_size;
    const float* pred = (const float*)d_in[0];
    const float* tgt  = (const float*)d_in[1];
    float* part = (float*)d_ws;
    float* out  = (float*)d_out;

    // nb must be a multiple of 64 (power of two) and fit in workspace.
    int nb = 2048;
    while ((size_t)6 * (size_t)nb * sizeof(float) > ws_size && nb > 64) nb >>= 1;

    ul_partial<<<nb, 256, 0, stream>>>(pred, tgt, part, nb);
    ul_final<<<1, 32, 0, stream>>>(part, out, nb);
}